// DiffusionDynamicOutput_79336635892254
// MI455X (gfx1250) — compile-verified
//
#include <hip/hip_runtime.h>

#define IN_CH  64
#define EMBED  1024
#define HID    256           // IN_CH * 4
#define KK     9             // 3x3 taps
#define KDIM   (IN_CH * KK)  // 576
#define BATCH  8
#define NTOK   32
#define HDIM   128
#define WDIM   128
#define WSTRIDE 577          // 576 padded by 1 float: kills 16-way LDS bank conflicts

typedef __attribute__((ext_vector_type(2))) float v2f;
typedef __attribute__((ext_vector_type(8))) float v8f;

// ---------------------------------------------------------------------------
// Kernel 1: h[bn, :] = silu( (wv_embs[bn] + t_emb[b]) @ W1 + b1 )   (256 x 256)
// ---------------------------------------------------------------------------
__global__ __launch_bounds__(HID) void mlp1_kernel(
    const float* __restrict__ t_emb, const float* __restrict__ wv,
    const float* __restrict__ W1, const float* __restrict__ b1,
    float* __restrict__ hout) {
  __shared__ float ctx[EMBED];
  const int bn  = blockIdx.x;        // b*NTOK + n
  const int b   = bn >> 5;           // NTOK == 32
  const int tid = threadIdx.x;       // output column 0..255
  for (int e = tid; e < EMBED; e += HID)
    ctx[e] = wv[(size_t)bn * EMBED + e] + t_emb[(size_t)b * EMBED + e];
  __syncthreads();
  float acc = b1[tid];
  for (int e = 0; e < EMBED; ++e)
    acc = fmaf(ctx[e], W1[(size_t)e * HID + tid], acc);   // W1 read coalesced
  const float sig = 1.0f / (1.0f + __expf(-acc));
  hout[(size_t)bn * HID + tid] = acc * sig;               // SiLU
}

// ---------------------------------------------------------------------------
// Kernel 2: wmat[bn, :] = h[bn] @ W2 + b2     (256 x 576)
// ---------------------------------------------------------------------------
__global__ __launch_bounds__(KDIM) void mlp2_kernel(
    const float* __restrict__ h, const float* __restrict__ W2,
    const float* __restrict__ b2, float* __restrict__ wmat) {
  __shared__ float hl[HID];
  const int bn  = blockIdx.x;
  const int tid = threadIdx.x;       // output column 0..575
  for (int k = tid; k < HID; k += KDIM) hl[k] = h[(size_t)bn * HID + k];
  __syncthreads();
  float acc = b2[tid];
  for (int k = 0; k < HID; ++k)
    acc = fmaf(hl[k], W2[(size_t)k * KDIM + tid], acc);   // W2 read coalesced
  wmat[(size_t)bn * KDIM + tid] = acc;
}

// ---------------------------------------------------------------------------
// Kernel 3: dynamic 3x3 depth-summed conv as 9 shift-GEMMs via f32 WMMA.
//   out[b,n,h,w] = sum_{d,kk} latent_pad[b,d,h+di-1,w+dj-1] * wmat[b,n,d*9+kk]
// Wave tile: 16 tokens (M) x 16 pixels (N); K=4 channels per WMMA; 144 WMMAs.
// Workgroup: 256 threads = 8 waves -> 32 tokens x 64 pixels of one image row.
// ---------------------------------------------------------------------------
__global__ __launch_bounds__(256) void dynconv_wmma_kernel(
    const float* __restrict__ latent, const float* __restrict__ wmat,
    float* __restrict__ out) {
  __shared__ float wlds[NTOK * WSTRIDE];   // 73,856 B of 320 KB LDS

  const int b   = blockIdx.z;
  const int h   = blockIdx.y;
  const int w0  = blockIdx.x * 64;
  const int tid = threadIdx.x;

  // Stage this batch's 32x576 dynamic weights into LDS (padded row stride).
  // Row-outer loop: no integer division, coalesced 576-float row reads.
  const float* wsrc = wmat + (size_t)b * NTOK * KDIM;
  for (int n = 0; n < NTOK; ++n) {
    for (int o = tid; o < KDIM; o += 256)
      wlds[n * WSTRIDE + o] = wsrc[n * KDIM + o];
  }
  __syncthreads();

  const int wave    = tid >> 5;
  const int lane    = tid & 31;
  const int m       = lane & 15;    // M row (A) / N col (B) within tile
  const int hiK     = lane >> 4;    // K-half select per ISA A/B layouts
  const int tokBase = (wave & 1) * 16;
  const int px0     = w0 + (wave >> 1) * 16;

  const float* lat_b = latent + (size_t)b * IN_CH * HDIM * WDIM;
  const float* arow  = &wlds[(tokBase + m) * WSTRIDE];

  v8f acc = {0.f, 0.f, 0.f, 0.f, 0.f, 0.f, 0.f, 0.f};

  for (int di = 0; di < 3; ++di) {
    const int  hh    = h + di - 1;
    const bool rowOK = (unsigned)hh < (unsigned)HDIM;
    for (int dj = 0; dj < 3; ++dj) {
      const int  kk = di * 3 + dj;
      const int  ww = px0 + m + dj - 1;
      const bool ok = rowOK && ((unsigned)ww < (unsigned)WDIM);
      const float* bbase = lat_b + (long)hh * WDIM + ww;  // deref only if ok
#pragma unroll
      for (int d = 0; d < IN_CH; d += 4) {
        const int c0 = d + 2 * hiK;
        v2f a, bv;
        // A: 16x4 weight tile, lane holds K = {c0, c0+1} of token tokBase+m
        a.x = arow[c0 * KK + kk];
        a.y = arow[(c0 + 1) * KK + kk];
        // B: 4x16 latent tile (shifted), zero outside the padded image
        bv.x = ok ? bbase[(size_t)c0 * (HDIM * WDIM)] : 0.0f;
        bv.y = ok ? bbase[(size_t)(c0 + 1) * (HDIM * WDIM)] : 0.0f;
        acc = __builtin_amdgcn_wmma_f32_16x16x4_f32(
            /*neg_a=*/false, a, /*neg_b=*/false, bv,
            /*c_mod=*/(short)0, acc, /*reuse_a=*/false, /*reuse_b=*/false);
      }
    }
  }

  // D tile store: VGPR r holds token tokBase + r + 8*hiK, pixel px0 + m.
  const int tok = tokBase + 8 * hiK;
#pragma unroll
  for (int r = 0; r < 8; ++r) {
    out[(((size_t)b * NTOK + tok + r) * HDIM + h) * WDIM + (px0 + m)] = acc[r];
  }
}

// ---------------------------------------------------------------------------
extern "C" void kernel_launch(void* const* d_in, const int* in_sizes, int n_in,
                              void* d_out, int out_size, void* d_ws, size_t ws_size,
                              hipStream_t stream) {
  const float* latent = (const float*)d_in[0];  // (8,64,128,128)
  const float* t_emb  = (const float*)d_in[1];  // (8,1024)
  const float* wv     = (const float*)d_in[2];  // (8,32,1024)
  const float* W1     = (const float*)d_in[3];  // (1024,256)
  const float* b1     = (const float*)d_in[4];  // (256)
  const float* W2     = (const float*)d_in[5];  // (256,576)
  const float* b2     = (const float*)d_in[6];  // (576)
  float* out = (float*)d_out;                   // (8,32,128,128)

  float* hbuf = (float*)d_ws;                       // 256*256 f32
  float* wmat = hbuf + (size_t)BATCH * NTOK * HID;  // 256*576 f32

  mlp1_kernel<<<BATCH * NTOK, HID, 0, stream>>>(t_emb, wv, W1, b1, hbuf);
  mlp2_kernel<<<BATCH * NTOK, KDIM, 0, stream>>>(hbuf, W2, b2, wmat);

  dim3 grid(WDIM / 64, HDIM, BATCH);
  dynconv_wmma_kernel<<<grid, 256, 0, stream>>>(latent, wmat, out);
}